// GCN_44805098832627
// MI455X (gfx1250) — compile-verified
//
#include <hip/hip_runtime.h>
#include <math.h>

typedef __attribute__((ext_vector_type(2))) float v2f;
typedef __attribute__((ext_vector_type(8))) float v8f;

#define HID 128
#define NGRAPH 64

// ---------------------------------------------------------------------------
// Generic fill
// ---------------------------------------------------------------------------
__global__ void fill_kernel(float* __restrict__ p, float v, long long n) {
    long long i = (long long)blockIdx.x * blockDim.x + threadIdx.x;
    if (i < n) p[i] = v;
}

// ---------------------------------------------------------------------------
// Degree accumulation over dst (deg pre-initialized to 1.0 for self-loops)
// ---------------------------------------------------------------------------
__global__ void deg_kernel(const int* __restrict__ ei, float* __restrict__ deg, int E) {
    int e = blockIdx.x * blockDim.x + threadIdx.x;
    if (e < E) atomicAdd(&deg[ei[E + e]], 1.0f);
}

__global__ void rsqrt_kernel(float* __restrict__ deg, int N) {
    int i = blockIdx.x * blockDim.x + threadIdx.x;
    if (i < N) {
        float d = deg[i];
        deg[i] = (d > 0.0f) ? rsqrtf(d) : 0.0f;
    }
}

// ---------------------------------------------------------------------------
// H = X @ W^T  via V_WMMA_F32_16X16X4_F32.
//   X: [N, K] row-major, W: [HID, K] row-major (torch Linear convention)
//   H: [N, HID]
// One wave computes a 16x16 tile; 8 waves/block cover all 128 output features
// of a 16-node slab. EXEC is all-ones everywhere a WMMA executes.
// A layout: lane L (half=L>>4) holds A[M=L&15][k0+2*half + {0,1}] in v[0..1]
// B layout: lane L holds B[k0+2*half + {0,1}][N=L&15] = W[col][k]
// D layout: vgpr r -> row = r + 8*half, col = L&15
// ---------------------------------------------------------------------------
template <int K>
__global__ __launch_bounds__(256) void linear_wmma(const float* __restrict__ X,
                                                   const float* __restrict__ W,
                                                   float* __restrict__ H, int N) {
    const int lane = threadIdx.x & 31;
    const int wv   = threadIdx.x >> 5;           // 0..7 -> output-feature tile
    const int rowBase = blockIdx.x * 16;
    const int colBase = wv * 16;

    const int half = lane >> 4;
    const int l15  = lane & 15;
    const int kk   = half * 2;

    int arow = rowBase + l15;
    if (arow > N - 1) arow = N - 1;              // clamp loads, keep EXEC full
    const float* __restrict__ xr = X + (long long)arow * K;
    const float* __restrict__ wr = W + (long long)(colBase + l15) * K;

    v8f acc = {};
#pragma unroll
    for (int k0 = 0; k0 < K; k0 += 4) {
        v2f a = *(const v2f*)(xr + k0 + kk);     // 8B aligned: kk even, k0%4==0
        v2f b = *(const v2f*)(wr + k0 + kk);
        acc = __builtin_amdgcn_wmma_f32_16x16x4_f32(
            /*neg_a=*/false, a, /*neg_b=*/false, b,
            /*c_mod=*/(short)0, acc, /*reuse_a=*/false, /*reuse_b=*/false);
    }

    float* __restrict__ out = H + (long long)(rowBase + half * 8) * HID + colBase + l15;
#pragma unroll
    for (int r = 0; r < 8; ++r) {
        if (rowBase + half * 8 + r < N) out[(long long)r * HID] = acc[r];
    }
}

// ---------------------------------------------------------------------------
// Edge scatter: one wave per (edge or self-loop), float4 gather of h[src],
// 4x global_atomic_add_f32 into agg[dst]. e >= E encodes self-loop node e-E.
// ---------------------------------------------------------------------------
__global__ __launch_bounds__(256) void scatter_kernel(const int* __restrict__ ei,
                                                      const float* __restrict__ dinv,
                                                      const float* __restrict__ h,
                                                      float* __restrict__ agg,
                                                      int E, int N) {
    long long gtid = (long long)blockIdx.x * blockDim.x + threadIdx.x;
    int e    = (int)(gtid >> 5);
    int lane = threadIdx.x & 31;
    if (e >= E + N) return;

    int s, d;
    if (e < E) { s = ei[e]; d = ei[E + e]; }
    else       { s = e - E; d = s; }

    float nrm = dinv[s] * dinv[d];
    const float4* __restrict__ hs = (const float4*)(h + (long long)s * HID);
    float* __restrict__ ad = agg + (long long)d * HID;

    float4 v = hs[lane];
    int c = lane * 4;
    atomicAdd(ad + c + 0, v.x * nrm);
    atomicAdd(ad + c + 1, v.y * nrm);
    atomicAdd(ad + c + 2, v.z * nrm);
    atomicAdd(ad + c + 3, v.w * nrm);
}

// ---------------------------------------------------------------------------
// a[i] = relu(a[i] + bias[i % 128])   (in place)
// ---------------------------------------------------------------------------
__global__ void bias_relu_kernel(float* __restrict__ a, const float* __restrict__ bias,
                                 long long total) {
    long long i = (long long)blockIdx.x * blockDim.x + threadIdx.x;
    if (i < total) {
        float v = a[i] + bias[i & (HID - 1)];
        a[i] = v > 0.0f ? v : 0.0f;
    }
}

// ---------------------------------------------------------------------------
// Global mean pool: per-(node,feature) atomic into pooled[batch[node]][f],
// per-node atomic count.
// ---------------------------------------------------------------------------
__global__ void pool_cnt_kernel(const int* __restrict__ batch, float* __restrict__ cnt,
                                int N) {
    int i = blockIdx.x * blockDim.x + threadIdx.x;
    if (i < N) atomicAdd(&cnt[batch[i]], 1.0f);
}

__global__ void pool_sum_kernel(const float* __restrict__ a, const int* __restrict__ batch,
                                float* __restrict__ pooled, int N) {
    long long i = (long long)blockIdx.x * blockDim.x + threadIdx.x;
    if (i < (long long)N * HID) {
        int node = (int)(i >> 7);
        int f    = (int)(i & (HID - 1));
        atomicAdd(&pooled[batch[node] * HID + f], a[i]);
    }
}

// ---------------------------------------------------------------------------
// out[g][o] = dot(sums[g], fc_w[o]) / max(cnt[g],1) + fc_b[o]
// ---------------------------------------------------------------------------
__global__ void fc_kernel(const float* __restrict__ pooled, const float* __restrict__ cnt,
                          const float* __restrict__ fw, const float* __restrict__ fb,
                          float* __restrict__ out, int OUTD) {
    int idx = blockIdx.x * blockDim.x + threadIdx.x;
    if (idx >= NGRAPH * OUTD) return;
    int g = idx / OUTD, o = idx % OUTD;
    float c = cnt[g];
    if (c < 1.0f) c = 1.0f;
    float s = 0.0f;
    for (int f = 0; f < HID; ++f) s += pooled[g * HID + f] * fw[o * HID + f];
    out[idx] = s / c + fb[o];
}

// ---------------------------------------------------------------------------
extern "C" void kernel_launch(void* const* d_in, const int* in_sizes, int n_in,
                              void* d_out, int out_size, void* d_ws, size_t ws_size,
                              hipStream_t stream) {
    const float* x     = (const float*)d_in[0];
    const int*   ei    = (const int*)d_in[1];
    const int*   batch = (const int*)d_in[2];
    const float* w1    = (const float*)d_in[3];
    const float* b1    = (const float*)d_in[4];
    const float* w2    = (const float*)d_in[5];
    const float* b2    = (const float*)d_in[6];
    const float* fw    = (const float*)d_in[7];
    const float* fb    = (const float*)d_in[8];
    float* out = (float*)d_out;

    const int N = in_sizes[2];          // 50000
    const int E = in_sizes[1] / 2;      // 800000
    const int OUTD = in_sizes[8];       // 10
    const long long NH = (long long)N * HID;

    // workspace layout (all float, offsets stay 16B aligned for float4)
    float* deg    = (float*)d_ws;       // N    (becomes dinv in place)
    float* cnt    = deg + N;            // 64
    float* pooled = cnt + 64;           // 64*128
    float* buf0   = pooled + NGRAPH * HID; // N*128 (linear output)
    float* buf1   = buf0 + NH;             // N*128 (aggregate / activation)

    const int TB = 256;
    dim3 blk(TB);
    int nBlkN    = (N + TB - 1) / TB;
    long long nBlkNH = (NH + TB - 1) / TB;
    int nBlkE    = (E + TB - 1) / TB;
    long long scatterThreads = (long long)(E + N) * 32;
    long long nBlkScatter = (scatterThreads + TB - 1) / TB;
    int mTiles = (N + 15) / 16;

    // --- normalization -----------------------------------------------------
    fill_kernel<<<nBlkN, blk, 0, stream>>>(deg, 1.0f, N);               // self-loop
    deg_kernel<<<nBlkE, blk, 0, stream>>>(ei, deg, E);
    rsqrt_kernel<<<nBlkN, blk, 0, stream>>>(deg, N);                    // deg -> dinv

    // --- layer 1: h = relu(Agg(x @ W1^T) + b1) -----------------------------
    linear_wmma<64><<<mTiles, blk, 0, stream>>>(x, w1, buf0, N);
    fill_kernel<<<(int)nBlkNH, blk, 0, stream>>>(buf1, 0.0f, NH);
    scatter_kernel<<<(int)nBlkScatter, blk, 0, stream>>>(ei, deg, buf0, buf1, E, N);
    bias_relu_kernel<<<(int)nBlkNH, blk, 0, stream>>>(buf1, b1, NH);

    // --- layer 2 -----------------------------------------------------------
    linear_wmma<128><<<mTiles, blk, 0, stream>>>(buf1, w2, buf0, N);
    fill_kernel<<<(int)nBlkNH, blk, 0, stream>>>(buf1, 0.0f, NH);       // buf1 consumed above
    scatter_kernel<<<(int)nBlkScatter, blk, 0, stream>>>(ei, deg, buf0, buf1, E, N);
    bias_relu_kernel<<<(int)nBlkNH, blk, 0, stream>>>(buf1, b2, NH);

    // --- global mean pool + FC --------------------------------------------
    fill_kernel<<<((64 + NGRAPH * HID) + TB - 1) / TB, blk, 0, stream>>>(cnt, 0.0f,
                                                                         64 + NGRAPH * HID);
    pool_cnt_kernel<<<nBlkN, blk, 0, stream>>>(batch, cnt, N);
    pool_sum_kernel<<<(int)nBlkNH, blk, 0, stream>>>(buf1, batch, pooled, N);
    fc_kernel<<<(NGRAPH * OUTD + TB - 1) / TB, blk, 0, stream>>>(pooled, cnt, fw, fb,
                                                                 out, OUTD);
}